// StreamingRhythmProjector_83038897700997
// MI455X (gfx1250) — compile-verified
//
#include <hip/hip_runtime.h>

// StreamingRhythmProjector for MI455X (gfx1250, wave32).
// Bandwidth-bound (~300MB HBM traffic): one 256-thread block per row,
// float4 (b128) global access, pass-1 intermediates staged in LDS,
// row reductions via V_WMMA_F32_16X16X4_F32 (f32 A/B -> exact f32 sums).

typedef __attribute__((ext_vector_type(2))) float v2f;
typedef __attribute__((ext_vector_type(8))) float v8f;

#define MAX_U    2048
#define NTHREADS 256
#define NWAVES   (NTHREADS / 32)

// Full-wave (32 lane) f32 sum using the CDNA5 WMMA f32 pipe.
// A (16x4 f32): a[0]=x -> A[m][0]=x_m (lanes 0-15), A[m][2]=x_{m+16}.
// B = all-ones => D[m][n] = x_m + x_{m+16}. Summing the 8 accumulator regs
// gives half-sums in lane halves; one shfl_xor(16) completes the wave sum.
__device__ __forceinline__ float wave_sum_f32(float x) {
#if __has_builtin(__builtin_amdgcn_wmma_f32_16x16x4_f32)
  v2f a; a[0] = x;    a[1] = 0.0f;
  v2f b; b[0] = 1.0f; b[1] = 1.0f;
  v8f c = {};
  c = __builtin_amdgcn_wmma_f32_16x16x4_f32(
      /*neg_a=*/false, a, /*neg_b=*/false, b,
      /*c_mod=*/(short)0, c, /*reuse_a=*/false, /*reuse_b=*/false);
  float s = ((c[0] + c[1]) + (c[2] + c[3])) + ((c[4] + c[5]) + (c[6] + c[7]));
  s += __shfl_xor(s, 16, 32);
  return s;
#else
  for (int off = 16; off > 0; off >>= 1) x += __shfl_xor(x, off, 32);
  return x;
#endif
}

#define PASS1_COMP(mm, aa, llr, ppw, bbl, SP, SC)  \
  {                                                \
    float anc_ = fmaxf((aa), 1.0f);                \
    float bse_ = anc_ * expf((llr));               \
    SP = fmaxf(fminf(bse_, anc_ * 3.0f), 1.0f) * (mm); \
    SC = fmaxf((ppw), 0.0f) * (0.5f + (bbl)) * (mm);   \
  }

__global__ __launch_bounds__(NTHREADS)
void StreamingRhythmProjector_kernel(
    const float* __restrict__ anchor_src,    // [B,U]
    const float* __restrict__ unit_mask,     // [B,U]
    const float* __restrict__ speech_budget, // [B,1]
    const float* __restrict__ pause_budget,  // [B,1]
    const float* __restrict__ logratio,      // [B,U]
    const float* __restrict__ pause_w,       // [B,U]
    const float* __restrict__ boundary,      // [B,U]
    const float* __restrict__ phase_ptr,     // [B]
    const float* __restrict__ backlog,       // [B]
    const float* __restrict__ clock_delta,   // [B]
    const int*   __restrict__ commit_prev,   // [B]
    const int*   __restrict__ open_run,      // [B,U]
    float* __restrict__ out_speech,          // [B,U]
    float* __restrict__ out_pause,           // [B,U]
    float* __restrict__ out_eff,             // [B,U]
    float* __restrict__ out_commit,          // [B] (int value stored as float)
    float* __restrict__ out_phase,           // [B]
    float* __restrict__ out_backlog,         // [B]
    float* __restrict__ out_clock,           // [B]
    int U) {
  __shared__ __align__(16) float s_sbase[MAX_U];   // masked speech base
  __shared__ __align__(16) float s_scores[MAX_U];  // masked pause scores
  __shared__ __align__(16) float s_anchor[MAX_U];  // raw dur_anchor_src
  __shared__ float s_part[3][NWAVES];
  __shared__ int   s_first_open;
  __shared__ int   s_commit;

  const int row  = blockIdx.x;
  const int tid  = threadIdx.x;
  const int lane = tid & 31;
  const int wave = tid >> 5;
  const size_t base = (size_t)row * (size_t)U;
  const int U4 = U >> 2;         // vectorized portion; tail handled scalar

  if (tid == 0) s_first_open = U;  // sentinel: no open run
  __syncthreads();

  // ---------------- Pass 1: elementwise + stage to LDS + partial sums -------
  const float4* an4 = (const float4*)(anchor_src + base);
  const float4* um4 = (const float4*)(unit_mask + base);
  const float4* lr4 = (const float4*)(logratio + base);
  const float4* pw4 = (const float4*)(pause_w + base);
  const float4* bl4 = (const float4*)(boundary + base);
  const int4*   op4 = (const int4*)(open_run + base);
  float4* sb4v = (float4*)s_sbase;
  float4* sc4v = (float4*)s_scores;
  float4* sa4v = (float4*)s_anchor;

  float acc_sbase = 0.0f, acc_score = 0.0f, acc_mask = 0.0f;
  for (int q = tid; q < U4; q += NTHREADS) {
    float4 m  = um4[q];
    float4 a  = an4[q];
    float4 lr = lr4[q];
    float4 pw = pw4[q];
    float4 bl = bl4[q];
    int4   op = op4[q];

    float4 sp, sc;
    PASS1_COMP(m.x, a.x, lr.x, pw.x, bl.x, sp.x, sc.x);
    PASS1_COMP(m.y, a.y, lr.y, pw.y, bl.y, sp.y, sc.y);
    PASS1_COMP(m.z, a.z, lr.z, pw.z, bl.z, sp.z, sc.z);
    PASS1_COMP(m.w, a.w, lr.w, pw.w, bl.w, sp.w, sc.w);

    sb4v[q] = sp;
    sc4v[q] = sc;
    sa4v[q] = a;

    acc_sbase += ((sp.x + sp.y) + (sp.z + sp.w));
    acc_score += ((sc.x + sc.y) + (sc.z + sc.w));
    acc_mask  += ((m.x + m.y) + (m.z + m.w));

    int i0 = q << 2, fo = U;
    if (op.w != 0 && m.w > 0.5f) fo = i0 + 3;
    if (op.z != 0 && m.z > 0.5f) fo = i0 + 2;
    if (op.y != 0 && m.y > 0.5f) fo = i0 + 1;
    if (op.x != 0 && m.x > 0.5f) fo = i0 + 0;
    if (fo < U) atomicMin(&s_first_open, fo);
  }
  for (int i = (U4 << 2) + tid; i < U; i += NTHREADS) {  // scalar tail
    float m = unit_mask[base + i], a = anchor_src[base + i];
    float sp, sc;
    PASS1_COMP(m, a, logratio[base + i], pause_w[base + i], boundary[base + i],
               sp, sc);
    s_sbase[i] = sp; s_scores[i] = sc; s_anchor[i] = a;
    acc_sbase += sp; acc_score += sc; acc_mask += m;
    if (open_run[base + i] != 0 && m > 0.5f) atomicMin(&s_first_open, i);
  }

  // Wave-level WMMA reductions (all lanes converged here).
  float w0 = wave_sum_f32(acc_sbase);
  float w1 = wave_sum_f32(acc_score);
  float w2 = wave_sum_f32(acc_mask);
  if (lane == 0) {
    s_part[0][wave] = w0;
    s_part[1][wave] = w1;
    s_part[2][wave] = w2;
  }
  __syncthreads();

  float tot_sbase = 0.0f, tot_score = 0.0f, tot_mask = 0.0f;
  for (int w = 0; w < NWAVES; ++w) {
    tot_sbase += s_part[0][w];
    tot_score += s_part[1][w];
    tot_mask  += s_part[2][w];
  }
  const int visible = (int)(tot_mask + 0.5f);  // prefix mask -> exact count

  // ---------------- Commit frontier (scalar, thread 0) ----------------------
  if (tid == 0) {
    int  first_open    = s_first_open;                 // gated on mask
    int  closed_prefix = (first_open < U) ? first_open : visible;
    int  release_cap   = max(visible - 2, 0);          // TAIL_HOLD_UNITS
    int  candidate     = min(release_cap, closed_prefix);
    int  prev          = commit_prev[row];
    int  b_idx         = min(max(candidate - 1, 0), U - 1);
    float b_val        = boundary[base + b_idx];
    bool soft = (candidate > 0) && (candidate < visible) && (b_val < 0.45f);
    if (soft) candidate = max(prev, candidate - 1);
    s_commit = max(prev, candidate);
  }
  __syncthreads();

  // ---------------- Pass 2: normalize, write outputs, segment sums ----------
  const int commit   = s_commit;
  const int prev     = commit_prev[row];
  const int prev_c   = min(max(prev, 0), U);     // JAX take_along_axis clamps
  const int commit_c = min(max(commit, 0), U);

  const float sb         = speech_budget[row];
  const float pb         = pause_budget[row];
  const float scale      = sb / fmaxf(tot_sbase, 1e-6f);
  const bool  use_scores = tot_score > 0.0f;
  const float inv_score  = 1.0f / fmaxf(tot_score, 1e-6f);
  const float fallback_w = 1.0f / fmaxf(tot_mask, 1.0f);

  float4* osp4 = (float4*)(out_speech + base);
  float4* opa4 = (float4*)(out_pause + base);
  float4* oef4 = (float4*)(out_eff + base);

  float acc_eff = 0.0f, acc_seg_eff = 0.0f, acc_seg_src = 0.0f;
  for (int q = tid; q < U4; q += NTHREADS) {
    const int i0 = q << 2;
    float4 sbv = sb4v[q];
    float4 scv = sc4v[q];
    float4 av  = sa4v[q];
    float4 sp, pa, ef;

#define PASS2_COMP(C, IDX)                                            \
    {                                                                 \
      float m_   = ((IDX) < visible) ? 1.0f : 0.0f;                   \
      sp.C       = sbv.C * scale * m_;                                \
      float wgt_ = use_scores ? (scv.C * inv_score) : (m_ * fallback_w); \
      pa.C       = wgt_ * pb;                                         \
      ef.C       = (sp.C + pa.C) * m_;                                \
      acc_eff += ef.C;                                                \
      if ((IDX) >= prev_c && (IDX) < commit_c) {                      \
        acc_seg_eff += ef.C;                                          \
        acc_seg_src += av.C;                                          \
      }                                                               \
    }
    PASS2_COMP(x, i0 + 0);
    PASS2_COMP(y, i0 + 1);
    PASS2_COMP(z, i0 + 2);
    PASS2_COMP(w, i0 + 3);
#undef PASS2_COMP

    osp4[q] = sp;
    opa4[q] = pa;
    oef4[q] = ef;
  }
  for (int i = (U4 << 2) + tid; i < U; i += NTHREADS) {  // scalar tail
    float m   = (i < visible) ? 1.0f : 0.0f;
    float sp  = s_sbase[i] * scale * m;
    float wgt = use_scores ? (s_scores[i] * inv_score) : (m * fallback_w);
    float pa  = wgt * pb;
    float ef  = (sp + pa) * m;
    out_speech[base + i] = sp;
    out_pause[base + i]  = pa;
    out_eff[base + i]    = ef;
    acc_eff += ef;
    if (i >= prev_c && i < commit_c) { acc_seg_eff += ef; acc_seg_src += s_anchor[i]; }
  }

  float e0 = wave_sum_f32(acc_eff);
  float e1 = wave_sum_f32(acc_seg_eff);
  float e2 = wave_sum_f32(acc_seg_src);
  if (lane == 0) {
    s_part[0][wave] = e0;
    s_part[1][wave] = e1;
    s_part[2][wave] = e2;
  }
  __syncthreads();

  if (tid == 0) {
    float eff_tot = 0.0f, seg_eff = 0.0f, seg_src = 0.0f;
    for (int w = 0; w < NWAVES; ++w) {
      eff_tot += s_part[0][w];
      seg_eff += s_part[1][w];
      seg_src += s_part[2][w];
    }
    const bool adv   = commit > prev;
    const float cd   = clock_delta[row];
    const float ncl  = adv ? (cd + (seg_eff - seg_src)) : cd;
    const float nbk  = adv ? fmaxf(ncl, 0.0f) : backlog[row];
    const float vtot = fmaxf(eff_tot, 1.0f);
    const float ph   = phase_ptr[row];
    const float nph  = adv ? fminf(fmaxf(ph + seg_eff / vtot, 0.0f), 1.0f) : ph;

    out_commit[row]  = (float)commit;
    out_phase[row]   = nph;
    out_backlog[row] = nbk;
    out_clock[row]   = ncl;
  }
}

extern "C" void kernel_launch(void* const* d_in, const int* in_sizes, int n_in,
                              void* d_out, int out_size, void* d_ws, size_t ws_size,
                              hipStream_t stream) {
  (void)n_in; (void)d_ws; (void)ws_size; (void)out_size;
  const int B = in_sizes[7];           // phase_ptr is [B]
  const int U = in_sizes[0] / B;       // dur_anchor_src is [B,U]

  float* out = (float*)d_out;
  const size_t BU = (size_t)B * (size_t)U;
  float* o_speech  = out;
  float* o_pause   = out + BU;
  float* o_eff     = out + 2 * BU;
  float* o_commit  = out + 3 * BU;
  float* o_phase   = o_commit + B;
  float* o_backlog = o_phase + B;
  float* o_clock   = o_backlog + B;

  StreamingRhythmProjector_kernel<<<B, NTHREADS, 0, stream>>>(
      (const float*)d_in[0],  (const float*)d_in[1],  (const float*)d_in[2],
      (const float*)d_in[3],  (const float*)d_in[4],  (const float*)d_in[5],
      (const float*)d_in[6],  (const float*)d_in[7],  (const float*)d_in[8],
      (const float*)d_in[9],  (const int*)d_in[10],   (const int*)d_in[11],
      o_speech, o_pause, o_eff, o_commit, o_phase, o_backlog, o_clock, U);
}